// VCRender_18726057411147
// MI455X (gfx1250) — compile-verified
//
#include <hip/hip_runtime.h>
#include <math.h>

#define H_ 128
#define W_ 128
#define MAXF 512
#define SIGMAINV_ 7000.0f
#define EPS_ 1e-10f

// squared distance from point (px,py) to segment (ax,ay)-(bx,by), matching reference _seg_d2
__device__ __forceinline__ float seg_d2(float px, float py, float ax, float ay,
                                        float bx, float by) {
    float abx = bx - ax, aby = by - ay;
    float apx = px - ax, apy = py - ay;
    float t = (apx * abx + apy * aby) / (abx * abx + aby * aby + EPS_);
    t = fminf(fmaxf(t, 0.0f), 1.0f);
    float dx = apx - t * abx;
    float dy = apy - t * aby;
    return dx * dx + dy * dy;
}

// Single-block per-face preprocessing with deterministic, ORDER-PRESERVING compaction
// of front&&valid faces (back/degenerate faces contribute exactly nothing to any
// output: inside==false and probf==0 => coverage factor exactly 1.0).
// Writes (compacted index p):
//   faceF[p*16] : {a0x,a0y,a0c, a1x,a1y,a1c, z0,z1,z2, x0,y0,x1,y1,x2,y2, pad}
//   colws[p*9]  : c0.rgb, c1.rgb, c2.rgb
//   *fcount     : number of kept faces
// Writes (original index f): outNormal[f*3] (3rd reference output, all faces).
__global__ void __launch_bounds__(MAXF)
face_setup_kernel(const float* __restrict__ pts,
                  const float* __restrict__ cols,
                  const float* __restrict__ rot,
                  const float* __restrict__ campos,
                  const float* __restrict__ proj,
                  const int*   __restrict__ faces,
                  float* __restrict__ faceF,
                  float* __restrict__ colws,
                  int*   __restrict__ fcount,
                  float* __restrict__ outNormal,
                  int F) {
    __shared__ int scan[MAXF];

    const int f = threadIdx.x;
    const bool active = f < F;

    float rec[16];
    float cw[9];
    int flag = 0;

    if (active) {
        int idx[3];
        idx[0] = faces[f * 3 + 0];
        idx[1] = faces[f * 3 + 1];
        idx[2] = faces[f * 3 + 2];

        float cx = campos[0], cy = campos[1], cz = campos[2];
        float prx = proj[0], pry = proj[1], prz = proj[2];

        float v[3][3];
        float x2d[3], y2d[3];
        for (int t = 0; t < 3; ++t) {
            float p0 = pts[idx[t] * 3 + 0] - cx;
            float p1 = pts[idx[t] * 3 + 1] - cy;
            float p2 = pts[idx[t] * 3 + 2] - cz;
            // cam[j] = sum_k p[k] * rot[j][k]   (einsum 'bpk,bjk->bpj')
            for (int j = 0; j < 3; ++j)
                v[t][j] = p0 * rot[j * 3 + 0] + p1 * rot[j * 3 + 1] + p2 * rot[j * 3 + 2];
            float zz = v[t][2] * prz;
            x2d[t] = (v[t][0] * prx) / zz;
            y2d[t] = (v[t][1] * pry) / zz;
        }

        float e1x = v[1][0] - v[0][0], e1y = v[1][1] - v[0][1], e1z = v[1][2] - v[0][2];
        float e2x = v[2][0] - v[0][0], e2y = v[2][1] - v[0][1], e2z = v[2][2] - v[0][2];
        float nx = e1y * e2z - e1z * e2y;
        float ny = e1z * e2x - e1x * e2z;
        float nz = e1x * e2y - e1y * e2x;
        float nlen = sqrtf(nx * nx + ny * ny + nz * nz + 1e-8f);
        float ninv = 1.0f / (nlen + 1e-15f);
        outNormal[f * 3 + 0] = nx * ninv;
        outNormal[f * 3 + 1] = ny * ninv;
        outNormal[f * 3 + 2] = nz * ninv;

        float x0 = x2d[0], y0 = y2d[0], x1 = x2d[1], y1 = y2d[1], x2 = x2d[2], y2 = y2d[2];
        float area = (x1 - x0) * (y2 - y0) - (x2 - x0) * (y1 - y0);
        bool valid = fabsf(area) > EPS_;
        float denom = valid ? area : EPS_;
        float rd = 1.0f / denom;

        bool front = nz > 0.0f;
        flag = (front && valid) ? 1 : 0;

        // w0 = ((x1-px)(y2-py)-(x2-px)(y1-py))/denom  -> affine in (px,py)
        rec[0] = (y1 - y2) * rd;
        rec[1] = (x2 - x1) * rd;
        rec[2] = (x1 * y2 - x2 * y1) * rd;
        // w1 = ((x2-px)(y0-py)-(x0-px)(y2-py))/denom
        rec[3] = (y2 - y0) * rd;
        rec[4] = (x0 - x2) * rd;
        rec[5] = (x2 * y0 - x0 * y2) * rd;
        rec[6] = v[0][2]; rec[7] = v[1][2]; rec[8] = v[2][2];
        rec[9] = x0;  rec[10] = y0;  rec[11] = x1;
        rec[12] = y1; rec[13] = x2;  rec[14] = y2;
        rec[15] = 0.0f;

        for (int t = 0; t < 3; ++t) {
            cw[t * 3 + 0] = cols[idx[t] * 3 + 0];
            cw[t * 3 + 1] = cols[idx[t] * 3 + 1];
            cw[t * 3 + 2] = cols[idx[t] * 3 + 2];
        }
    }

    // deterministic inclusive Hillis-Steele scan over the keep-flags
    scan[f] = flag;
    __syncthreads();
    for (int off = 1; off < MAXF; off <<= 1) {
        int vv = scan[f];
        if (f >= off) vv += scan[f - off];
        __syncthreads();
        scan[f] = vv;
        __syncthreads();
    }

    if (flag) {
        int pos = scan[f] - 1;                 // exclusive position
        float* fd = faceF + (size_t)pos * 16;
        #pragma unroll
        for (int i = 0; i < 16; ++i) fd[i] = rec[i];
        float* cd = colws + (size_t)pos * 9;
        #pragma unroll
        for (int i = 0; i < 9; ++i) cd[i] = cw[i];
    }
    if (f == MAXF - 1) *fcount = scan[MAXF - 1];
}

// One thread per pixel. Compacted face table staged into LDS with the gfx1250
// async global->LDS DMA path (ASYNCcnt).
__global__ void __launch_bounds__(256)
raster_kernel(const float* __restrict__ faceF,
              const float* __restrict__ colws,
              const int*   __restrict__ fcount,
              float* __restrict__ out) {
    __shared__ float4 sm4[MAXF * 4];   // up to 32 KB of the 320 KB WGP LDS

    const int tid = threadIdx.x;
    const int Fc = *fcount;            // uniform -> scalar load

    // ---- async stage Fc*64 bytes into LDS ----
    {
        unsigned lds0 = (unsigned)(uintptr_t)(&sm4[0]);   // LDS aperture: low 32 bits = LDS offset
        int nchunks = Fc * 4;                             // 16B chunks
        for (int c = tid; c < nchunks; c += 256) {
            unsigned ldsoff = lds0 + (unsigned)c * 16u;
            unsigned long long gaddr =
                (unsigned long long)(uintptr_t)(faceF + (size_t)c * 4);
            asm volatile("global_load_async_to_lds_b128 %0, %1, off"
                         :
                         : "v"(ldsoff), "v"(gaddr)
                         : "memory");
        }
        asm volatile("s_wait_asynccnt 0x0" ::: "memory");
    }
    __syncthreads();

    const int pix = blockIdx.x * blockDim.x + tid;
    if (pix >= H_ * W_) return;
    const int row = pix / W_;
    const int col = pix - row * W_;
    const float px = ((float)col + 0.5f) / (float)W_ * 2.0f - 1.0f;
    const float py = 1.0f - ((float)row + 0.5f) / (float)H_ * 2.0f;

    float bestz = __builtin_inff();
    int   bestf = -1;
    float bw0 = 0.0f, bw1 = 0.0f, bw2 = 0.0f;
    float prod = 1.0f;   // prod over faces of (1 - probf)

    for (int f = 0; f < Fc; ++f) {
        const float4* fd4 = &sm4[f * 4];
        float4 A = fd4[0];   // a0x a0y a0c a1x
        float4 B = fd4[1];   // a1y a1c z0  z1
        float4 C = fd4[2];   // z2  x0  y0  x1
        float4 D = fd4[3];   // y1  x2  y2  pad

        float w0 = fmaf(A.x, px, fmaf(A.y, py, A.z));
        float w1 = fmaf(A.w, px, fmaf(B.x, py, B.y));
        float w2 = 1.0f - w0 - w1;
        bool inside = (w0 >= 0.0f) & (w1 >= 0.0f) & (w2 >= 0.0f);

        float z = w0 * B.z + w1 * B.w + w2 * C.x;
        if (inside && (z < bestz)) {            // strict <  == first-argmin semantics
            bestz = z; bestf = f;
            bw0 = w0; bw1 = w1; bw2 = w2;
        }

        float x0 = C.y, y0 = C.z, x1 = C.w, y1 = D.x, x2 = D.y, y2 = D.z;
        float d2 = fminf(fminf(seg_d2(px, py, x0, y0, x1, y1),
                               seg_d2(px, py, x1, y1, x2, y2)),
                         seg_d2(px, py, x2, y2, x0, y0));
        if (inside) d2 = 0.0f;
        prod *= (1.0f - expf(-d2 * SIGMAINV_));
    }

    float r = 0.0f, g = 0.0f, b = 0.0f;
    if (bestf >= 0) {
        const float* cw = colws + (size_t)bestf * 9;
        r = bw0 * cw[0] + bw1 * cw[3] + bw2 * cw[6];
        g = bw0 * cw[1] + bw1 * cw[4] + bw2 * cw[7];
        b = bw0 * cw[2] + bw1 * cw[5] + bw2 * cw[8];
    }

    // outputs: imrender (H*W*3) | improb (H*W) | normal1 (F*3, written by face_setup)
    out[pix * 3 + 0] = r;
    out[pix * 3 + 1] = g;
    out[pix * 3 + 2] = b;
    out[H_ * W_ * 3 + pix] = 1.0f - prod;
}

extern "C" void kernel_launch(void* const* d_in, const int* in_sizes, int n_in,
                              void* d_out, int out_size, void* d_ws, size_t ws_size,
                              hipStream_t stream) {
    const float* pts    = (const float*)d_in[0];  // (1,256,3)
    const float* cols   = (const float*)d_in[1];  // (1,256,3)
    const float* rot    = (const float*)d_in[2];  // (1,3,3)
    const float* campos = (const float*)d_in[3];  // (1,3)
    const float* proj   = (const float*)d_in[4];  // (3,1)
    const int*   faces  = (const int*)d_in[5];    // (512,3)

    int F = in_sizes[5] / 3;
    if (F > MAXF) F = MAXF;

    float* out       = (float*)d_out;
    float* faceF     = (float*)d_ws;                         // MAXF*16 floats
    float* colws     = faceF + (size_t)MAXF * 16;            // MAXF*9 floats
    int*   fcount    = (int*)(colws + (size_t)MAXF * 9);     // 1 int
    float* outNormal = out + (size_t)H_ * W_ * 3 + (size_t)H_ * W_;

    face_setup_kernel<<<1, MAXF, 0, stream>>>(
        pts, cols, rot, campos, proj, faces, faceF, colws, fcount, outNormal, F);

    raster_kernel<<<(H_ * W_ + 255) / 256, 256, 0, stream>>>(faceF, colws, fcount, out);
}